// GCN_47871705481343
// MI455X (gfx1250) — compile-verified
//
#include <hip/hip_runtime.h>
#include <math.h>

// ---------------------------------------------------------------------------
// GATv2 x3 on MI455X (gfx1250, wave32).
// Projections via V_WMMA_F32_16X16X4_F32 (exact f32 semantics; GEMM flops are
// negligible, problem is L2-resident gather/scatter bound: xl/xr (25.6MB) and
// the output accumulator (12.8MB) live in the 192MB L2, so per-edge gathers
// and scatter atomicAdds resolve in L2, not HBM).
// Edge projection (E x 13 @ 13 x 64) is fused as register-resident WMMA tiles
// inside the attention-logit kernel -> no 205MB intermediate buffer.
// ---------------------------------------------------------------------------

typedef __attribute__((ext_vector_type(2))) float v2f;
typedef __attribute__((ext_vector_type(8))) float v8f;

#define GAT_N 50000
#define GAT_E 800000
#define GAT_HID 64
#define NSLOPE 0.2f

__device__ __forceinline__ v8f wmma_f32_16x16x4(v2f a, v2f b, v8f c) {
  // (neg_a, A, neg_b, B, c_mod, C, reuse_a, reuse_b)
  return __builtin_amdgcn_wmma_f32_16x16x4_f32(false, a, false, b, (short)0, c,
                                               false, false);
}

// Order-preserving float<->uint encoding so float-max == uint-max.
__device__ __forceinline__ unsigned f2ord(float f) {
  unsigned u = __float_as_uint(f);
  return (u & 0x80000000u) ? ~u : (u | 0x80000000u);
}
__device__ __forceinline__ float ord2f(unsigned v) {
  return (v & 0x80000000u) ? __uint_as_float(v & 0x7fffffffu)
                           : __uint_as_float(~v);
}

// ---------------------------------------------------------------------------
// Node projections: XL = X @ Wl^T + bl ; XR = X @ Wr^T + br
// grid.x = N/16 tiles; 256 threads = 8 waves: wid = colTile(4) x which(2).
// One wave owns one 16x16 output tile. DIN is a template constant (12 or 64,
// both multiples of 4) -> fully unrolled WMMA chain, unconditional aligned
// b64 loads, no EXEC masking anywhere near the WMMAs.
// ---------------------------------------------------------------------------
template <int DIN>
__global__ void __launch_bounds__(256) gat_node_proj(
    const float* __restrict__ X, const float* __restrict__ Wl,
    const float* __restrict__ Bl, const float* __restrict__ Wr,
    const float* __restrict__ Br, float* __restrict__ XL,
    float* __restrict__ XR) {
  static_assert(DIN % 4 == 0, "DIN must be a multiple of 4");
  int wid = threadIdx.x >> 5;
  int lane = threadIdx.x & 31;
  int colT = wid & 3;
  int which = wid >> 2;
  int tileM = blockIdx.x;

  const float* W = which ? Wr : Wl;
  const float* Bv = which ? Br : Bl;
  float* O = which ? XR : XL;

  int hi = lane >> 4;  // half-wave selector (K pair)
  int lo = lane & 15;  // M (for A) / N (for B,C)
  int row = tileM * 16 + lo;
  int col = colT * 16 + lo;

  const float* arow = X + (size_t)row * DIN;
  const float* brow = W + (size_t)col * DIN;

  v8f c = {};
#pragma unroll
  for (int k0 = 0; k0 < DIN; k0 += 4) {
    int ka = k0 + hi * 2;  // A/B layout: v0 holds K=k0+2*hi, v1 holds +1
    v2f a = *(const v2f*)(arow + ka);  // 8B-aligned: row*DIN and ka are even
    v2f b = *(const v2f*)(brow + ka);
    c = wmma_f32_16x16x4(a, b, c);
  }

  float bv = Bv[col];
#pragma unroll
  for (int r = 0; r < 8; ++r) {
    int orow = tileM * 16 + hi * 8 + r;  // C/D: M = r + 8*(lane>=16)
    O[(size_t)orow * GAT_HID + col] = c[r] + bv;
  }
}

// ---------------------------------------------------------------------------
// Per-layer init: zero accumulator [N*64], seg-max := -inf (encoded), denom:=0.
// ---------------------------------------------------------------------------
__global__ void gat_init_layer(unsigned* __restrict__ SMAXU,
                               float* __restrict__ DEN,
                               float* __restrict__ ACC) {
  int i = blockIdx.x * blockDim.x + threadIdx.x;
  if (i < GAT_N * GAT_HID) ACC[i] = 0.f;
  if (i < GAT_N) {
    SMAXU[i] = 0x007fffffu;  // f2ord(-inf)
    DEN[i] = 0.f;
  }
}

// ---------------------------------------------------------------------------
// Fused: e-tile = edge_attr @ We^T via WMMA (registers only), then
// s[e] = sum_h leakyrelu(xl[src]+xr[dst]+e)_h * att_h, segment-max via
// encoded atomicMax. One wave handles 16 edges; grid.x = (E/16)/8, block 256.
// K=13: steps 0,4,8 are unconditional; the K=12..15 tail is branchless
// (load K=12 unconditionally, zero the dead lanes via cndmask on hi).
// ---------------------------------------------------------------------------
__global__ void __launch_bounds__(256) gat_edge_logits(
    const float* __restrict__ EA,   // [E,13]
    const float* __restrict__ WE,   // [64,13]
    const float* __restrict__ ATT,  // [64]
    const int* __restrict__ src, const int* __restrict__ dst,
    const float* __restrict__ XL, const float* __restrict__ XR,
    float* __restrict__ S, unsigned* __restrict__ SMAXU) {
  int wid = threadIdx.x >> 5;
  int lane = threadIdx.x & 31;
  int tile = blockIdx.x * 8 + wid;  // E/16 = 50000, grid = 6250 -> exact
  int hi = lane >> 4;
  int lo = lane & 15;
  int row = tile * 16 + lo;  // edge row for A loads

  const float* earow = EA + (size_t)row * 13;
  float ea12 = earow[12];  // K=12 element, shared across column tiles

  // e-tile: 16 edges x 64 features, 4 column tiles, K = 13 padded to 16
  v8f acc[4] = {};
#pragma unroll
  for (int t = 0; t < 4; ++t) {
    int col = t * 16 + lo;
    const float* werow = WE + (size_t)col * 13;
#pragma unroll
    for (int k0 = 0; k0 < 12; k0 += 4) {  // ka+1 <= 11 < 13: always in range
      int ka = k0 + hi * 2;
      v2f a, b;
      a.x = earow[ka];
      a.y = earow[ka + 1];
      b.x = werow[ka];
      b.y = werow[ka + 1];
      acc[t] = wmma_f32_16x16x4(a, b, acc[t]);
    }
    {  // tail: only K=12 exists; hi==1 lanes and .y are zero padding
      float we12 = werow[12];
      v2f a = {}, b = {};
      a.x = hi ? 0.f : ea12;
      b.x = hi ? 0.f : we12;
      acc[t] = wmma_f32_16x16x4(a, b, acc[t]);
    }
  }

  // Hoisted attention-vector gather (4 scalars per lane).
  float attv[4];
#pragma unroll
  for (int t = 0; t < 4; ++t) attv[t] = ATT[t * 16 + lo];

  // Logits: lane holds, for 8 edges (M = hi*8+r), column t*16+lo of e-tile.
  float sp[8];
#pragma unroll
  for (int r = 0; r < 8; ++r) {
    int em = tile * 16 + hi * 8 + r;
    const float* xls = XL + (size_t)src[em] * GAT_HID;
    const float* xrd = XR + (size_t)dst[em] * GAT_HID;
    float p = 0.f;
#pragma unroll
    for (int t = 0; t < 4; ++t) {
      int col = t * 16 + lo;
      float m = xls[col] + xrd[col] + acc[t][r];
      m = (m > 0.f) ? m : NSLOPE * m;
      p += m * attv[t];
    }
    // reduce over the 16 lanes of this half-wave (xor stays within halves)
    p += __shfl_xor(p, 1);
    p += __shfl_xor(p, 2);
    p += __shfl_xor(p, 4);
    p += __shfl_xor(p, 8);
    sp[r] = p;
  }
  if (lo == 0) {  // lanes 0 and 16 each own 8 edges
#pragma unroll
    for (int r = 0; r < 8; ++r) {
      int em = tile * 16 + hi * 8 + r;
      S[em] = sp[r];
      atomicMax(&SMAXU[dst[em]], f2ord(sp[r]));
    }
  }
}

// ---------------------------------------------------------------------------
// ex = exp(s - segmax[dst]); denom[dst] += ex  (L2-resident atomics)
// ---------------------------------------------------------------------------
__global__ void gat_expsum(const float* __restrict__ S,
                           const unsigned* __restrict__ SMAXU,
                           const int* __restrict__ dst,
                           float* __restrict__ EXB, float* __restrict__ DEN) {
  int e = blockIdx.x * blockDim.x + threadIdx.x;
  if (e >= GAT_E) return;
  int d = dst[e];
  float ex = __expf(S[e] - ord2f(SMAXU[d]));
  EXB[e] = ex;
  atomicAdd(&DEN[d], ex);
}

// ---------------------------------------------------------------------------
// ACC[dst] += xl[src] * alpha ; one thread per (edge, feature).
// ---------------------------------------------------------------------------
__global__ void gat_aggregate(const float* __restrict__ EXB,
                              const float* __restrict__ DEN,
                              const float* __restrict__ XL,
                              const int* __restrict__ src,
                              const int* __restrict__ dst,
                              float* __restrict__ ACC) {
  long long idx = (long long)blockIdx.x * blockDim.x + threadIdx.x;
  if (idx >= (long long)GAT_E * GAT_HID) return;
  int e = (int)(idx >> 6);
  int h = (int)(idx & 63);
  int d = dst[e];
  float alpha = EXB[e] / (DEN[d] + 1e-16f);
  atomicAdd(&ACC[(size_t)d * GAT_HID + h],
            XL[(size_t)src[e] * GAT_HID + h] * alpha);
}

// ---------------------------------------------------------------------------
// out += bias ; optional ReLU (layers 0,1)
// ---------------------------------------------------------------------------
__global__ void gat_bias_act(float* __restrict__ ACC,
                             const float* __restrict__ BIAS, int do_relu) {
  int i = blockIdx.x * blockDim.x + threadIdx.x;
  if (i >= GAT_N * GAT_HID) return;
  float v = ACC[i] + BIAS[i & 63];
  ACC[i] = do_relu ? fmaxf(v, 0.f) : v;
}

// ---------------------------------------------------------------------------
extern "C" void kernel_launch(void* const* d_in, const int* in_sizes, int n_in,
                              void* d_out, int out_size, void* d_ws,
                              size_t ws_size, hipStream_t stream) {
  (void)in_sizes; (void)n_in; (void)out_size; (void)ws_size;

  const int N = GAT_N, E = GAT_E;

  const float* x = (const float*)d_in[0];
  const int* eidx = (const int*)d_in[1];
  const float* ea = (const float*)d_in[2];
  const int* src = eidx;
  const int* dst = eidx + E;

  // Workspace carve (~45 MB, 256B aligned slices)
  char* ws = (char*)d_ws;
  auto carve = [&](size_t bytes) {
    char* p = ws;
    ws += (bytes + 255) & ~(size_t)255;
    return p;
  };
  float* XL = (float*)carve((size_t)N * GAT_HID * 4);
  float* XR = (float*)carve((size_t)N * GAT_HID * 4);
  float* H = (float*)carve((size_t)N * GAT_HID * 4);  // inter-layer buffer
  float* S = (float*)carve((size_t)E * 4);
  float* EXB = (float*)carve((size_t)E * 4);
  unsigned* SMAXU = (unsigned*)carve((size_t)N * 4);
  float* DEN = (float*)carve((size_t)N * 4);

  float* out = (float*)d_out;

  const float* cur_in = x;
  for (int l = 0; l < 3; ++l) {
    const int base = 3 + l * 7;
    const float* wl = (const float*)d_in[base + 0];
    const float* bl = (const float*)d_in[base + 1];
    const float* wr = (const float*)d_in[base + 2];
    const float* br = (const float*)d_in[base + 3];
    const float* we = (const float*)d_in[base + 4];
    const float* att = (const float*)d_in[base + 5];
    const float* bias = (const float*)d_in[base + 6];
    float* target = (l == 2) ? out : H;

    // N/16 = 3125 tiles, 8 waves/block (4 col tiles x {Wl,Wr})
    if (l == 0)
      gat_node_proj<12><<<N / 16, 256, 0, stream>>>(cur_in, wl, bl, wr, br,
                                                    XL, XR);
    else
      gat_node_proj<64><<<N / 16, 256, 0, stream>>>(cur_in, wl, bl, wr, br,
                                                    XL, XR);
    gat_init_layer<<<(N * GAT_HID + 255) / 256, 256, 0, stream>>>(SMAXU, DEN,
                                                                  target);
    // (E/16)/8 = 6250 blocks, 8 waves each, 16 edges per wave (exact)
    gat_edge_logits<<<(E / 16) / 8, 256, 0, stream>>>(ea, we, att, src, dst,
                                                      XL, XR, S, SMAXU);
    gat_expsum<<<(E + 255) / 256, 256, 0, stream>>>(S, SMAXU, dst, EXB, DEN);
    gat_aggregate<<<(int)(((long long)E * GAT_HID) / 256), 256, 0, stream>>>(
        EXB, DEN, XL, src, dst, target);
    gat_bias_act<<<(N * GAT_HID + 255) / 256, 256, 0, stream>>>(target, bias,
                                                                l < 2 ? 1 : 0);
    cur_in = H;
  }
}